// GroupedGaborFilterbank_32384053412454
// MI455X (gfx1250) — compile-verified
//
#include <hip/hip_runtime.h>
#include <hip/hip_bf16.h>

typedef float v2f __attribute__((ext_vector_type(2)));
typedef float v8f __attribute__((ext_vector_type(8)));

#define TIME_LEN 64000
#define OUT_T 400

// Workspace float offsets.
// kern stored transposed [32][KPS], KPS = KP+2: cos ch in cols 0..9, sin ch in cols
// 16..25, zeros elsewhere (incl. rows >= K).
//   g0: KPS=290 @0 (9280), g1: KPS=146 @9280 (4672), g2: KPS=74 @13952 (2368),
//   g3: KPS=46 @16320 (1472); end 17792
// win [WS][10]: g0 @17792 (410), g1 @18202 (1010), g2 @19212 (2010), g3 @21222 (4010);
//   end 25232

__global__ void gabor_setup(const float* __restrict__ cf, const float* __restrict__ bw,
                            const float* __restrict__ pw, float* __restrict__ ws) {
  const int g = blockIdx.x;
  const int CS[4] = {10, 4, 2, 1};
  const int K [4] = {287, 143, 71, 41};
  const int KPS[4] = {290, 146, 74, 46};
  const int WS[4] = {41, 101, 201, 401};
  const int KO[4] = {0, 9280, 13952, 16320};
  const int WO[4] = {17792, 18202, 19212, 21222};
  const int a = g * 10;
  const int cs = CS[g], k = K[g], kps = KPS[g], wsz = WS[g];
  float* kern = ws + KO[g];
  float* win  = ws + WO[g];
  const float PI = 3.14159265358979323846f;
  const float Z  = 0.37473527f;            // sqrt(2 ln2)/pi
  const float RSQRT2PI = 0.3989422804f;    // 1/sqrt(2*pi)

  for (int idx = threadIdx.x; idx < 32 * kps; idx += blockDim.x) {
    const int col = idx / kps, row = idx % kps;  // transposed: [col][row]
    float v = 0.0f;
    const bool is_cos = (col < 10);
    const bool is_sin = (col >= 16 && col < 26);
    if (row < k && (is_cos || is_sin)) {
      const int ch = is_cos ? col : (col - 16);
      const float t = (float)(row - k / 2);
      const float c = fminf(fmaxf(cf[a + ch], 0.0f), PI);
      const float b = fminf(fmaxf(bw[a + ch], 2.0f * Z), 401.0f * Z);
      const float gauss = expf(-(t * t) / (2.0f * b * b));
      const float norm  = RSQRT2PI / b;
      const float ph = t * c;
      v = norm * gauss * (is_cos ? cosf(ph) : sinf(ph));
    }
    kern[idx] = v;
  }
  for (int idx = threadIdx.x; idx < wsz * 10; idx += blockDim.x) {
    const int j = idx / 10, ch = idx % 10;
    const float p = fminf(fmaxf(pw[a + ch], 2.0f / 401.0f), 0.5f);
    const float sigma = p / (float)cs * 401.0f / (float)wsz;
    const float u = ((float)j - 0.5f * (float)(wsz - 1)) / (sigma * 0.5f * (float)(wsz - 1));
    win[idx] = expf(-0.5f * u * u);
  }
}

// One workgroup = (batch b, TP pooled outputs) for one filter group.
// Phase 1: stage signal window + transposed zero-padded kernel in LDS.
// Phase 2: conv via V_WMMA_F32_16X16X4_F32 (im2col GEMM, M=16 positions,
//          N=16 channels, K=taps). B fragments are contiguous v2f (ds_load_b64
//          straight into even-aligned VGPR pairs). Envelope = ccos^2+csin^2 -> LDS.
// Phase 3: segmented Gaussian pooling from LDS -> global out (32,400,40).
template <int CS, int PS, int K, int KP, int WS, int TP, int KOFF, int WOFF, int LCONV, int CHOFF>
__global__ __launch_bounds__(256) void gabor_group(const float* __restrict__ x,
                                                   const float* __restrict__ wsbuf,
                                                   float* __restrict__ out) {
  constexpr int KPS  = KP + 2;                  // transposed kernel column stride
  constexpr int NC   = (TP - 1) * PS + WS;      // conv outputs needed by this tile
  constexpr int NT   = (NC + 15) / 16;          // 16-row WMMA tiles
  constexpr int NR   = NT * 16;                 // padded conv rows in LDS
  constexpr int XLP  = (NR - 1) * CS + KP + 2;  // padded signal window (floats)
  constexpr int NOUT = TP * 10;                 // pooled outputs per workgroup
  constexpr int NSEG = (256 / NOUT) > 1 ? (256 / NOUT) : 1;  // pooling segments

  __shared__ float x_lds[XLP];
  __shared__ float o_lds[NR * 10];
  __shared__ __align__(16) float kern_lds[32 * KPS];
  __shared__ float win_lds[WS * 10];
  __shared__ float psum[NSEG > 1 ? NOUT * NSEG : 1];

  const int b  = blockIdx.y;
  const int t0 = blockIdx.x * TP;              // first pooled output index
  const int p0 = t0 * PS - WS / 2;             // first conv position (may be <0)
  const long xbeg = (long)p0 * CS - K / 2;     // first signal sample (may be <0)
  const float* xb = x + (long)b * TIME_LEN;

  for (int i = threadIdx.x; i < 32 * KPS; i += 256) kern_lds[i] = wsbuf[KOFF + i];
  for (int i = threadIdx.x; i < WS * 10; i += 256) win_lds[i] = wsbuf[WOFF + i];
  for (int i = threadIdx.x; i < NR * 10; i += 256) o_lds[i] = 0.0f;
  for (int i = threadIdx.x; i < XLP; i += 256) {
    const long gx = xbeg + i;
    x_lds[i] = (gx >= 0 && gx < TIME_LEN) ? xb[gx] : 0.0f;
  }
  __syncthreads();

  const int lane = threadIdx.x & 31;
  const int wave = threadIdx.x >> 5;
  const int n    = lane & 15;        // M index for A, N index for B/C
  const int half = lane >> 4;        // 0: K rows 0/1, 1: K rows 2/3
  const int koff = half * 2;

  // Workgroup-uniform: all NR conv rows are valid positions (true for ~96% of blocks).
  const bool interior = (p0 >= 0) && (p0 + NR <= LCONV);

  // Transposed-kernel base pointers: fragment (rows kb+koff, kb+koff+1) of column n
  // is a contiguous, 8B-aligned v2f.
  const float* pkc = kern_lds + n * KPS + koff;         // cos column
  const float* pks = kern_lds + (16 + n) * KPS + koff;  // sin column

  for (int tile = wave; tile < NT; tile += 8) {  // uniform per wave -> EXEC all 1s
    const int ptile = tile * 16;
    v8f ccos = {};
    v8f csin = {};
    const float* px = x_lds + (ptile + n) * CS + koff;  // + kb
#pragma unroll 4
    for (int kb = 0; kb < KP; kb += 4) {
      v2f A;
      A.x = px[kb];
      A.y = px[kb + 1];
      const v2f Bc = *(const v2f*)(pkc + kb);
      const v2f Bs = *(const v2f*)(pks + kb);
      ccos = __builtin_amdgcn_wmma_f32_16x16x4_f32(false, A, false, Bc, (short)0, ccos,
                                                   false, false);
      csin = __builtin_amdgcn_wmma_f32_16x16x4_f32(false, A, false, Bs, (short)0, csin,
                                                   false, false);
    }
    // C layout: vgpr i -> row (i + 8*half), col n. Envelope is element-wise.
    float* orow = o_lds + (ptile + half * 8) * 10 + n;
    if (interior) {
      if (n < 10) {   // single lane-constant predicate for all 8 stores
#pragma unroll
        for (int i = 0; i < 8; ++i) {
          orow[i * 10] = ccos[i] * ccos[i] + csin[i] * csin[i];
        }
      }
    } else {
#pragma unroll
      for (int i = 0; i < 8; ++i) {
        const int pabs = p0 + ptile + half * 8 + i;
        if (n < 10 && pabs >= 0 && pabs < LCONV) {
          orow[i * 10] = ccos[i] * ccos[i] + csin[i] * csin[i];
        }
      }
    }
  }
  __syncthreads();

  if constexpr (NSEG == 1) {
    for (int tid = threadIdx.x; tid < NOUT; tid += 256) {
      const int dt = tid / 10, ch = tid % 10;
      float acc = 0.0f;
      const int base = dt * PS * 10 + ch;
      for (int j = 0; j < WS; ++j) acc += win_lds[j * 10 + ch] * o_lds[base + j * 10];
      out[((long)b * OUT_T + t0 + dt) * 40 + CHOFF + ch] = acc;
    }
  } else {
    // Split the pooling window into NSEG segments to use more lanes, then reduce.
    constexpr int JSEG = (WS + NSEG - 1) / NSEG;
    if (threadIdx.x < NOUT * NSEG) {
      const int o   = threadIdx.x % NOUT;
      const int seg = threadIdx.x / NOUT;
      const int dt = o / 10, ch = o % 10;
      const int j0 = seg * JSEG;
      const int j1 = (j0 + JSEG < WS) ? (j0 + JSEG) : WS;
      float acc = 0.0f;
      const int base = dt * PS * 10 + ch;
      for (int j = j0; j < j1; ++j) acc += win_lds[j * 10 + ch] * o_lds[base + j * 10];
      psum[seg * NOUT + o] = acc;
    }
    __syncthreads();
    if (threadIdx.x < NOUT) {
      const int dt = threadIdx.x / 10, ch = threadIdx.x % 10;
      float acc = 0.0f;
#pragma unroll
      for (int s = 0; s < NSEG; ++s) acc += psum[s * NOUT + threadIdx.x];
      out[((long)b * OUT_T + t0 + dt) * 40 + CHOFF + ch] = acc;
    }
  }
}

extern "C" void kernel_launch(void* const* d_in, const int* in_sizes, int n_in,
                              void* d_out, int out_size, void* d_ws, size_t ws_size,
                              hipStream_t stream) {
  const float* x  = (const float*)d_in[0];
  const float* cf = (const float*)d_in[1];
  const float* bw = (const float*)d_in[2];
  const float* pw = (const float*)d_in[3];
  float* out = (float*)d_out;
  float* wsf = (float*)d_ws;

  gabor_setup<<<4, 256, 0, stream>>>(cf, bw, pw, wsf);

  //            CS  PS   K   KP   WS  TP  KOFF   WOFF   LCONV  CHOFF
  gabor_group<  10, 16, 287, 288,  41, 16,     0, 17792,  6400,  0>
      <<<dim3(25, 32), 256, 0, stream>>>(x, wsf, out);
  gabor_group<   4, 40, 143, 144, 101, 16,  9280, 18202, 16000, 10>
      <<<dim3(25, 32), 256, 0, stream>>>(x, wsf, out);
  gabor_group<   2, 80,  71,  72, 201, 16, 13952, 19212, 32000, 20>
      <<<dim3(25, 32), 256, 0, stream>>>(x, wsf, out);
  gabor_group<   1,160,  41,  44, 401,  8, 16320, 21222, 64000, 30>
      <<<dim3(50, 32), 256, 0, stream>>>(x, wsf, out);
}